// TangentNonLin_6390911336495
// MI455X (gfx1250) — compile-verified
//
#include <hip/hip_runtime.h>

// modReLU on complex data, MI455X (gfx1250, wave32).
//
// Bandwidth-bound: 25.6M complex elems, 16 B traffic each -> 409.6 MB,
// floor ~17.6 us at 23.3 TB/s. Strategy:
//   - float4 (b128) vectorized loads/stores, fully coalesced
//   - non-temporal hints on the 410 MB stream (working set > 192 MB L2),
//     regular (cached) loads only for the 512 B bias
//   - one v_rsq_f32 per element instead of sqrt + divide:
//       s   = re^2 + im^2
//       inv = rsq(s); r = s*inv          (== sqrt(s) to ~1 ulp)
//       scale = relu(r + b) * inv
//       out = (s > 0) ? scale * x : x    (branchless; kills rsq(0) NaN)
// No matrix op exists in this workload, so WMMA does not apply.

typedef float float4v __attribute__((ext_vector_type(4)));

__global__ __launch_bounds__(256) void modrelu_f32_kernel(
    const float4v* __restrict__ x_re,
    const float4v* __restrict__ x_im,
    const float4v* __restrict__ bias,   // 32 float4s = 128 channels
    float4v* __restrict__ out_re,       // d_out            (N*C floats)
    float4v* __restrict__ out_im,       // d_out + N*C      (N*C floats)
    int n4)                             // N*C/4
{
    const int i = blockIdx.x * blockDim.x + threadIdx.x;
    if (i >= n4) return;

    // Streamed once: non-temporal loads (global_load_b128 th:NT).
    float4v re = __builtin_nontemporal_load(&x_re[i]);
    float4v im = __builtin_nontemporal_load(&x_im[i]);

    // Bias is tiny and reused by every row: regular cached load.
    // 4 consecutive channels per thread -> bias float4 index = i mod (128/4).
    float4v b = bias[i & 31];

    float4v ore, oim;
#pragma unroll
    for (int k = 0; k < 4; ++k) {
        float rk = re[k];
        float ik = im[k];
        float s  = __builtin_fmaf(rk, rk, ik * ik);       // |x|^2
        float inv = __builtin_amdgcn_rsqf(s);             // v_rsq_f32
        float r   = s * inv;                              // ~= sqrt(s)
        float scale = fmaxf(r + b[k], 0.0f) * inv;        // relu(r+b)/r
        bool  m = (s > 0.0f);                             // origin mask
        ore[k] = m ? scale * rk : rk;                     // v_cndmask
        oim[k] = m ? scale * ik : ik;
    }

    // Streamed once: non-temporal stores (global_store_b128 th:NT).
    __builtin_nontemporal_store(ore, &out_re[i]);
    __builtin_nontemporal_store(oim, &out_im[i]);
}

extern "C" void kernel_launch(void* const* d_in, const int* in_sizes, int n_in,
                              void* d_out, int out_size, void* d_ws, size_t ws_size,
                              hipStream_t stream) {
    // setup_inputs() order: x_re [N,C] f32, x_im [N,C] f32, bias [1,C] f32.
    const float* x_re = (const float*)d_in[0];
    const float* x_im = (const float*)d_in[1];
    const float* bias = (const float*)d_in[2];
    float*       out  = (float*)d_out;          // [2, N, C]: re block then im block

    const long long nc = (long long)in_sizes[0];  // N*C = 25,600,000 (divisible by 4)
    const int n4 = (int)(nc / 4);

    const int threads = 256;                      // 8 wave32s per block
    const int blocks  = (n4 + threads - 1) / threads;

    modrelu_f32_kernel<<<blocks, threads, 0, stream>>>(
        (const float4v*)x_re,
        (const float4v*)x_im,
        (const float4v*)bias,
        (float4v*)out,
        (float4v*)(out + nc),
        n4);
}